// Mamba2_4372276707432
// MI455X (gfx1250) — compile-verified
//
#include <hip/hip_runtime.h>
#include <hip/hip_bf16.h>

// ---------------- problem constants (fixed by the reference) ----------------
constexpr int kInner   = 2048;
constexpr int kState   = 128;
constexpr int kHead    = 32;
constexpr int kNConv   = 4;
constexpr int kChunk   = 64;
constexpr int kEmbd    = 1024;
constexpr int kHeadDim = kInner / kHead;            // 64
constexpr int kConvDim = kInner + 2 * kState;       // 2304
constexpr int kDIn     = 2 * kInner + 2 * kState + kHead; // 4384
constexpr int kBatch   = 2;
constexpr int kSeq     = 4096;
constexpr int kNChunk  = kSeq / kChunk;             // 64
constexpr int kRows    = kBatch * kSeq;             // 8192

typedef __attribute__((ext_vector_type(2))) float v2f;
typedef __attribute__((ext_vector_type(8))) float v8f;

__device__ __forceinline__ v8f wmma4(v2f a, v2f b, v8f c) {
  // D = A(16x4,f32) * B(4x16,f32) + C(16x16,f32)  -- exact fp32 matrix op
  return __builtin_amdgcn_wmma_f32_16x16x4_f32(false, a, false, b, (short)0, c,
                                               false, false);
}
__device__ __forceinline__ float sigmoid_(float x) {
  return 1.0f / (1.0f + expf(-x));
}

// ---------------------------------------------------------------------------
// Generic fp32 WMMA GEMM: C[M,N] = A[M,K] @ B[K,N], row major.
// One wave computes a 64x32 tile (8 wmma accumulators): 6 loads per 8 WMMAs,
// doubling register-level B reuse vs a 32x32 tile. Wave-uniform guard keeps
// EXEC all-ones for WMMA. Requires M % 64 == 0, N % 32 == 0, K % 4 == 0.
// ---------------------------------------------------------------------------
__global__ __launch_bounds__(256) void k_gemm_f32(
    const float* __restrict__ A, const float* __restrict__ B,
    float* __restrict__ C, int M, int N, int K) {
  const int lane = threadIdx.x & 31;
  const int wave = threadIdx.x >> 5;
  const int nt = N >> 5;
  const int total = (M >> 6) * nt;
  const int tile = blockIdx.x * 8 + wave;
  if (tile >= total) return;                 // wave-uniform: EXEC stays all-1s
  const int m0 = (tile / nt) << 6;
  const int n0 = (tile % nt) << 5;
  const int r  = lane & 15;
  const int kq = (lane >> 4) << 1;           // 0 or 2
  const float* Ar0 = A + (size_t)(m0 + r) * K;
  const float* Ar1 = Ar0 + (size_t)16 * K;
  const float* Ar2 = Ar0 + (size_t)32 * K;
  const float* Ar3 = Ar0 + (size_t)48 * K;
  v8f acc[4][2] = {};
  for (int k0 = 0; k0 < K; k0 += 4) {
    const int kk = k0 + kq;
    v2f a0, a1, a2, a3, b0, b1;
    a0.x = Ar0[kk]; a0.y = Ar0[kk + 1];
    a1.x = Ar1[kk]; a1.y = Ar1[kk + 1];
    a2.x = Ar2[kk]; a2.y = Ar2[kk + 1];
    a3.x = Ar3[kk]; a3.y = Ar3[kk + 1];
    const float* Bp = B + (size_t)kk * N + n0 + r;
    b0.x = Bp[0];  b0.y = Bp[N];
    b1.x = Bp[16]; b1.y = Bp[N + 16];
    acc[0][0] = wmma4(a0, b0, acc[0][0]);
    acc[0][1] = wmma4(a0, b1, acc[0][1]);
    acc[1][0] = wmma4(a1, b0, acc[1][0]);
    acc[1][1] = wmma4(a1, b1, acc[1][1]);
    acc[2][0] = wmma4(a2, b0, acc[2][0]);
    acc[2][1] = wmma4(a2, b1, acc[2][1]);
    acc[3][0] = wmma4(a3, b0, acc[3][0]);
    acc[3][1] = wmma4(a3, b1, acc[3][1]);
  }
  const int rb = (lane >> 4) << 3;
#pragma unroll
  for (int q = 0; q < 4; ++q) {
#pragma unroll
    for (int v = 0; v < 8; ++v) {
      float* Cp = C + (size_t)(m0 + 16 * q + rb + v) * N + n0 + r;
      Cp[0] = acc[q][0][v];
      Cp[16] = acc[q][1][v];
    }
  }
}

// ---------------------------------------------------------------------------
// Depthwise causal conv (4 taps) + SiLU over the xBC slice of zxbcdt.
// xBC_out[b,l,ch] layout with ch fastest (coalesced).
// ---------------------------------------------------------------------------
__global__ __launch_bounds__(256) void k_conv_silu(
    const float* __restrict__ zxbcdt, const float* __restrict__ init_conv,
    const float* __restrict__ conv_w, const float* __restrict__ conv_b,
    float* __restrict__ xBC) {
  const size_t idx = (size_t)blockIdx.x * 256 + threadIdx.x;
  if (idx >= (size_t)kBatch * kSeq * kConvDim) return;
  const int ch = (int)(idx % kConvDim);
  const int l  = (int)((idx / kConvDim) % kSeq);
  const int b  = (int)(idx / ((size_t)kConvDim * kSeq));
  float acc = conv_b[ch];
#pragma unroll
  for (int j = 0; j < kNConv; ++j) {
    const int t = l + j - (kNConv - 1);
    const float xv = (t >= 0)
        ? zxbcdt[((size_t)b * kSeq + t) * kDIn + kInner + ch]
        : init_conv[((size_t)b * kConvDim + ch) * kNConv + (kNConv + t)];
    acc += conv_w[ch * kNConv + j] * xv;
  }
  xBC[idx] = acc * sigmoid_(acc);
}

// new_conv_state[b,ch,j] = raw x_in at l = L-4+j  (L >= N_CONV)
__global__ __launch_bounds__(256) void k_conv_state(
    const float* __restrict__ zxbcdt, float* __restrict__ out_state) {
  const int idx = blockIdx.x * 256 + threadIdx.x;
  if (idx >= kBatch * kConvDim * kNConv) return;
  const int j  = idx % kNConv;
  const int ch = (idx / kNConv) % kConvDim;
  const int b  = idx / (kNConv * kConvDim);
  out_state[idx] =
      zxbcdt[((size_t)b * kSeq + (kSeq - kNConv + j)) * kDIn + kInner + ch];
}

// dt[b,l,h] = softplus(zxbcdt[..., D_IN-32+h] + dt_bias[h])
__global__ __launch_bounds__(256) void k_dt(
    const float* __restrict__ zxbcdt, const float* __restrict__ dt_bias,
    float* __restrict__ dt) {
  const int idx = blockIdx.x * 256 + threadIdx.x;
  if (idx >= kRows * kHead) return;
  const int h = idx % kHead;
  const size_t row = (size_t)(idx / kHead);
  const float x = zxbcdt[row * kDIn + (kDIn - kHead) + h] + dt_bias[h];
  dt[idx] = (x > 20.f) ? x : log1pf(expf(x));
}

// ---------------------------------------------------------------------------
// Per-(b,chunk,head) SSM intra-chunk work, all contractions on WMMA.
//   G = Cc @ Bc^T (K=128), mask by exp(segsum), Y_diag = M @ xc (K=64),
//   states = xc^T @ (decay*Bc) (K=64).  LDS: 49.7 KB.
// ---------------------------------------------------------------------------
__global__ __launch_bounds__(128) void k_ssm_chunk(
    const float* __restrict__ xBC, const float* __restrict__ dt,
    const float* __restrict__ A_log, float* __restrict__ ybuf,
    float* __restrict__ states, float* __restrict__ acs) {
  __shared__ float sAcs[kChunk];
  __shared__ float sDt[kChunk];
  __shared__ float sXc[kChunk * kHeadDim];   // xc[l][p]
  __shared__ float sBM[kChunk * kState];     // Bc[l][n], later masked M[l][s]
  const int h = blockIdx.x % kHead;
  const int c = (blockIdx.x / kHead) % kNChunk;
  const int b = blockIdx.x / (kHead * kNChunk);
  const int tid = threadIdx.x;
  const int lane = tid & 31, wave = tid >> 5;
  const size_t row0 = (size_t)b * kSeq + (size_t)c * kChunk;
  const float Ah = -expf(A_log[h]);

  if (tid < kChunk) {
    const float d = dt[(row0 + tid) * kHead + h];
    sDt[tid] = d;
    sAcs[tid] = Ah * d;
  }
  __syncthreads();
  if (tid == 0) {                            // tiny serial inclusive cumsum
    float s = 0.f;
    for (int i = 0; i < kChunk; ++i) { s += sAcs[i]; sAcs[i] = s; }
  }
  __syncthreads();
  if (tid < kChunk)
    acs[(((size_t)b * kHead + h) * kNChunk + c) * kChunk + tid] = sAcs[tid];
  for (int i = tid; i < kChunk * kHeadDim; i += 128) {
    const int l = i >> 6, p = i & 63;
    sXc[i] = xBC[(row0 + l) * kConvDim + h * kHeadDim + p] * sDt[l];
  }
  for (int i = tid; i < kChunk * kState; i += 128) {
    const int l = i >> 7, n = i & 127;
    sBM[i] = xBC[(row0 + l) * kConvDim + kInner + n];
  }
  __syncthreads();

  const int r  = lane & 15;
  const int kq = (lane >> 4) << 1;
  const int rb = (lane >> 4) << 3;
  const int gm = (wave >> 1) << 5;   // row (l) quadrant offset
  const int gs = (wave & 1) << 5;    // col (s / p) quadrant offset

  // ---- G = Cc @ Bc^T, masked, kept in registers ----
  v8f g00 = {}, g01 = {}, g10 = {}, g11 = {};
  {
    const float* C0 = xBC + (row0 + gm + r) * kConvDim + kInner + kState;
    const float* C1 = C0 + (size_t)16 * kConvDim;
    for (int k0 = 0; k0 < kState; k0 += 4) {
      const int kk = k0 + kq;
      v2f a0, a1, b0, b1;
      a0.x = C0[kk]; a0.y = C0[kk + 1];
      a1.x = C1[kk]; a1.y = C1[kk + 1];
      b0.x = sBM[(gs + r) * kState + kk];      b0.y = sBM[(gs + r) * kState + kk + 1];
      b1.x = sBM[(gs + 16 + r) * kState + kk]; b1.y = sBM[(gs + 16 + r) * kState + kk + 1];
      g00 = wmma4(a0, b0, g00); g01 = wmma4(a0, b1, g01);
      g10 = wmma4(a1, b0, g10); g11 = wmma4(a1, b1, g11);
    }
  }
#pragma unroll
  for (int v = 0; v < 8; ++v) {        // L-mask: exp(Acs[l]-Acs[s]) for s<=l
    const int r0 = gm + rb + v, r1 = r0 + 16;
    const int s0 = gs + r, s1 = s0 + 16;
    g00[v] = (s0 <= r0) ? g00[v] * expf(sAcs[r0] - sAcs[s0]) : 0.f;
    g01[v] = (s1 <= r0) ? g01[v] * expf(sAcs[r0] - sAcs[s1]) : 0.f;
    g10[v] = (s0 <= r1) ? g10[v] * expf(sAcs[r1] - sAcs[s0]) : 0.f;
    g11[v] = (s1 <= r1) ? g11[v] * expf(sAcs[r1] - sAcs[s1]) : 0.f;
  }
  __syncthreads();                     // all waves done with raw Bc
  const float alast = sAcs[kChunk - 1];
  for (int i = tid; i < kChunk * kState; i += 128) {
    const int l = i >> 7;
    sBM[i] *= expf(alast - sAcs[l]);   // decay_states * Bc
  }
  __syncthreads();

  // ---- states[p][n] = sum_l xc[l][p] * (decay*Bc)[l][n]  (64x128) ----
  const size_t sbase =
      ((((size_t)b * kNChunk + c) * kHead + h) * kHeadDim) * kState;
#pragma unroll
  for (int qq = 0; qq < 2; ++qq) {
    const int q = wave + 4 * qq;
    const int p0 = (q >> 2) << 5, n0 = (q & 3) << 5;
    v8f s00 = {}, s01 = {}, s10 = {}, s11 = {};
    for (int k0 = 0; k0 < kChunk; k0 += 4) {
      const int kk = k0 + kq;
      v2f a0, a1, b0, b1;
      a0.x = sXc[kk * kHeadDim + p0 + r];      a0.y = sXc[(kk + 1) * kHeadDim + p0 + r];
      a1.x = sXc[kk * kHeadDim + p0 + 16 + r]; a1.y = sXc[(kk + 1) * kHeadDim + p0 + 16 + r];
      b0.x = sBM[kk * kState + n0 + r];        b0.y = sBM[(kk + 1) * kState + n0 + r];
      b1.x = sBM[kk * kState + n0 + 16 + r];   b1.y = sBM[(kk + 1) * kState + n0 + 16 + r];
      s00 = wmma4(a0, b0, s00); s01 = wmma4(a0, b1, s01);
      s10 = wmma4(a1, b0, s10); s11 = wmma4(a1, b1, s11);
    }
#pragma unroll
    for (int v = 0; v < 8; ++v) {
      float* S0 = states + sbase + (size_t)(p0 + rb + v) * kState + n0 + r;
      float* S1 = states + sbase + (size_t)(p0 + 16 + rb + v) * kState + n0 + r;
      S0[0] = s00[v]; S0[16] = s01[v];
      S1[0] = s10[v]; S1[16] = s11[v];
    }
  }
  __syncthreads();                     // done with decayed Bc; reuse as M
#pragma unroll
  for (int v = 0; v < 8; ++v) {
    const int r0 = gm + rb + v, r1 = r0 + 16;
    sBM[r0 * kChunk + gs + r]      = g00[v];
    sBM[r0 * kChunk + gs + 16 + r] = g01[v];
    sBM[r1 * kChunk + gs + r]      = g10[v];
    sBM[r1 * kChunk + gs + 16 + r] = g11[v];
  }
  __syncthreads();

  // ---- Y_diag = M @ xc (64x64) ----
  {
    v8f y00 = {}, y01 = {}, y10 = {}, y11 = {};
    for (int k0 = 0; k0 < kChunk; k0 += 4) {
      const int kk = k0 + kq;
      v2f a0, a1, b0, b1;
      a0.x = sBM[(gm + r) * kChunk + kk];      a0.y = sBM[(gm + r) * kChunk + kk + 1];
      a1.x = sBM[(gm + 16 + r) * kChunk + kk]; a1.y = sBM[(gm + 16 + r) * kChunk + kk + 1];
      b0.x = sXc[kk * kHeadDim + gs + r];      b0.y = sXc[(kk + 1) * kHeadDim + gs + r];
      b1.x = sXc[kk * kHeadDim + gs + 16 + r]; b1.y = sXc[(kk + 1) * kHeadDim + gs + 16 + r];
      y00 = wmma4(a0, b0, y00); y01 = wmma4(a0, b1, y01);
      y10 = wmma4(a1, b0, y10); y11 = wmma4(a1, b1, y11);
    }
#pragma unroll
    for (int v = 0; v < 8; ++v) {
      float* Y0 = ybuf + (row0 + gm + rb + v) * kInner + h * kHeadDim + gs + r;
      float* Y1 = ybuf + (row0 + gm + 16 + rb + v) * kInner + h * kHeadDim + gs + r;
      Y0[0] = y00[v]; Y0[16] = y01[v];
      Y1[0] = y10[v]; Y1[16] = y11[v];
    }
  }
}

// ---------------------------------------------------------------------------
// Inter-chunk state recurrence: N_0 = init, N_{c+1} = exp(al_c)*N_c + st_c.
// Overwrites states[b,c,h] with the *previous* (pre-chunk) state N_c,
// writes N_64 to final_state.
// ---------------------------------------------------------------------------
__global__ __launch_bounds__(256) void k_scan(
    const float* __restrict__ init_ssm, const float* __restrict__ acs,
    float* __restrict__ states, float* __restrict__ final_out) {
  const int h = blockIdx.x % kHead;
  const int b = blockIdx.x / kHead;
  const int tid = threadIdx.x;           // each thread owns 32 of 8192 elems
  float S[32];
  const size_t ibase =
      ((size_t)b * kHead + h) * kHeadDim * kState + (size_t)tid * 32;
#pragma unroll
  for (int i = 0; i < 32; ++i) S[i] = init_ssm[ibase + i];
  for (int c = 0; c < kNChunk; ++c) {
    const float w = expf(
        acs[(((size_t)b * kHead + h) * kNChunk + c) * kChunk + (kChunk - 1)]);
    const size_t sb =
        ((((size_t)b * kNChunk + c) * kHead + h) * kHeadDim) * kState +
        (size_t)tid * 32;
#pragma unroll
    for (int i = 0; i < 32; ++i) {
      const float st = states[sb + i];
      states[sb + i] = S[i];            // prev-state consumed by Y_off
      S[i] = w * S[i] + st;
    }
  }
#pragma unroll
  for (int i = 0; i < 32; ++i) final_out[ibase + i] = S[i];
}

// ---------------------------------------------------------------------------
// Y_off[l,p] = exp(Acs[l]) * (Cc @ prevS^T)[l,p];  y += Y_off + x*D[h]
// ---------------------------------------------------------------------------
__global__ __launch_bounds__(128) void k_yoff(
    const float* __restrict__ xBC, const float* __restrict__ acs,
    const float* __restrict__ states, const float* __restrict__ Dv,
    float* __restrict__ ybuf) {
  __shared__ float sS[kHeadDim * kState];
  __shared__ float sAcs[kChunk];
  const int h = blockIdx.x % kHead;
  const int c = (blockIdx.x / kHead) % kNChunk;
  const int b = blockIdx.x / (kHead * kNChunk);
  const int tid = threadIdx.x;
  const int lane = tid & 31, wave = tid >> 5;
  const size_t row0 = (size_t)b * kSeq + (size_t)c * kChunk;
  const size_t sbase =
      ((((size_t)b * kNChunk + c) * kHead + h) * kHeadDim) * kState;
  for (int i = tid; i < kHeadDim * kState; i += 128) sS[i] = states[sbase + i];
  if (tid < kChunk)
    sAcs[tid] = acs[(((size_t)b * kHead + h) * kNChunk + c) * kChunk + tid];
  __syncthreads();
  const int r = lane & 15, kq = (lane >> 4) << 1, rb = (lane >> 4) << 3;
  const int m0 = (wave >> 1) << 5;  // l quadrant
  const int p0 = (wave & 1) << 5;   // p quadrant
  v8f y00 = {}, y01 = {}, y10 = {}, y11 = {};
  const float* C0 = xBC + (row0 + m0 + r) * kConvDim + kInner + kState;
  const float* C1 = C0 + (size_t)16 * kConvDim;
  for (int k0 = 0; k0 < kState; k0 += 4) {
    const int kk = k0 + kq;
    v2f a0, a1, b0, b1;
    a0.x = C0[kk]; a0.y = C0[kk + 1];
    a1.x = C1[kk]; a1.y = C1[kk + 1];
    b0.x = sS[(p0 + r) * kState + kk];      b0.y = sS[(p0 + r) * kState + kk + 1];
    b1.x = sS[(p0 + 16 + r) * kState + kk]; b1.y = sS[(p0 + 16 + r) * kState + kk + 1];
    y00 = wmma4(a0, b0, y00); y01 = wmma4(a0, b1, y01);
    y10 = wmma4(a1, b0, y10); y11 = wmma4(a1, b1, y11);
  }
  const float Dh = Dv[h];
#pragma unroll
  for (int v = 0; v < 8; ++v) {
    const int l0 = m0 + rb + v, l1 = l0 + 16;
    const float e0 = expf(sAcs[l0]), e1 = expf(sAcs[l1]);
    const int pA = p0 + r, pB = p0 + 16 + r;
    const size_t yr0 = (row0 + l0) * kInner + h * kHeadDim;
    const size_t yr1 = (row0 + l1) * kInner + h * kHeadDim;
    const size_t xr0 = (row0 + l0) * kConvDim + h * kHeadDim;
    const size_t xr1 = (row0 + l1) * kConvDim + h * kHeadDim;
    ybuf[yr0 + pA] += y00[v] * e0 + xBC[xr0 + pA] * Dh;
    ybuf[yr0 + pB] += y01[v] * e0 + xBC[xr0 + pB] * Dh;
    ybuf[yr1 + pA] += y10[v] * e1 + xBC[xr1 + pA] * Dh;
    ybuf[yr1 + pB] += y11[v] * e1 + xBC[xr1 + pB] * Dh;
  }
}

// ---------------------------------------------------------------------------
// y *= silu(z); RMSNorm over 2048; * norm_w  (in place on ybuf)
// ---------------------------------------------------------------------------
__global__ __launch_bounds__(256) void k_gatenorm(
    const float* __restrict__ zxbcdt, const float* __restrict__ norm_w,
    float* __restrict__ ybuf) {
  __shared__ float red[256];
  const size_t row = blockIdx.x;
  const int tid = threadIdx.x;
  float vals[8];
  float ss = 0.f;
#pragma unroll
  for (int j = 0; j < 8; ++j) {
    const int i = tid + j * 256;
    float y = ybuf[row * kInner + i];
    const float z = zxbcdt[row * kDIn + i];
    y *= z * sigmoid_(z);
    vals[j] = y;
    ss += y * y;
  }
  red[tid] = ss;
  __syncthreads();
  for (int s = 128; s > 0; s >>= 1) {
    if (tid < s) red[tid] += red[tid + s];
    __syncthreads();
  }
  const float scale = rsqrtf(red[0] * (1.0f / kInner) + 1e-5f);
#pragma unroll
  for (int j = 0; j < 8; ++j) {
    const int i = tid + j * 256;
    ybuf[row * kInner + i] = vals[j] * scale * norm_w[i];
  }
}

// ---------------------------------------------------------------------------
extern "C" void kernel_launch(void* const* d_in, const int* in_sizes, int n_in,
                              void* d_out, int out_size, void* d_ws,
                              size_t ws_size, hipStream_t stream) {
  const float* u         = (const float*)d_in[0];
  const float* init_conv = (const float*)d_in[1];
  const float* init_ssm  = (const float*)d_in[2];
  const float* w_in      = (const float*)d_in[3];
  const float* conv_w    = (const float*)d_in[4];
  const float* conv_b    = (const float*)d_in[5];
  const float* dt_bias   = (const float*)d_in[6];
  const float* A_log     = (const float*)d_in[7];
  const float* Dv        = (const float*)d_in[8];
  const float* norm_w    = (const float*)d_in[9];
  const float* w_out     = (const float*)d_in[10];

  float* out = (float*)d_out;
  float* out_y     = out;                                    // 8388608
  float* out_conv  = out + (size_t)kRows * kEmbd;            // +18432
  float* out_final = out_conv + (size_t)kBatch * kConvDim * kNConv;

  // workspace layout (floats); total ~105.6M floats (~423 MB)
  float* ws = (float*)d_ws;
  float* zxbcdt = ws;                                        // 8192*4384
  float* xBC    = zxbcdt + (size_t)kRows * kDIn;             // 8192*2304
  float* dtb    = xBC + (size_t)kRows * kConvDim;            // 8192*32
  float* acsb   = dtb + (size_t)kRows * kHead;               // 2*32*64*64
  float* states = acsb + (size_t)kBatch * kHead * kNChunk * kChunk; // 33.5M
  float* ybuf   = states +
      (size_t)kBatch * kNChunk * kHead * kHeadDim * kState;  // 8192*2048

  // 1) zxbcdt = u @ w_in   (8192 x 4384 x K=1024)
  {
    const int tiles = (kRows / 64) * (kDIn / 32);
    k_gemm_f32<<<(tiles + 7) / 8, 256, 0, stream>>>(u, w_in, zxbcdt, kRows,
                                                    kDIn, kEmbd);
  }
  // 2) depthwise conv + SiLU, conv state, dt
  {
    const size_t n = (size_t)kBatch * kSeq * kConvDim;
    k_conv_silu<<<(unsigned)((n + 255) / 256), 256, 0, stream>>>(
        zxbcdt, init_conv, conv_w, conv_b, xBC);
    k_conv_state<<<(kBatch * kConvDim * kNConv + 255) / 256, 256, 0, stream>>>(
        zxbcdt, out_conv);
    k_dt<<<(kRows * kHead + 255) / 256, 256, 0, stream>>>(zxbcdt, dt_bias, dtb);
  }
  // 3) intra-chunk SSM (Y_diag + per-chunk states)
  k_ssm_chunk<<<kBatch * kNChunk * kHead, 128, 0, stream>>>(
      xBC, dtb, A_log, ybuf, states, acsb);
  // 4) inter-chunk recurrence (+ final_state output)
  k_scan<<<kBatch * kHead, 256, 0, stream>>>(init_ssm, acsb, states, out_final);
  // 5) Y_off + x*D accumulation
  k_yoff<<<kBatch * kNChunk * kHead, 128, 0, stream>>>(xBC, acsb, states, Dv,
                                                       ybuf);
  // 6) gate with silu(z) + RMSNorm
  k_gatenorm<<<kRows, 256, 0, stream>>>(zxbcdt, norm_w, ybuf);
  // 7) out_y = ybuf @ w_out  (8192 x 1024 x K=2048)
  {
    const int tiles = (kRows / 64) * (kEmbd / 32);
    k_gemm_f32<<<(tiles + 7) / 8, 256, 0, stream>>>(ybuf, w_out, out_y, kRows,
                                                    kEmbd, kInner);
  }
}